// GeometryLocationAttentionHead_76845554860755
// MI455X (gfx1250) — compile-verified
//
#include <hip/hip_runtime.h>
#include <math.h>

typedef __attribute__((ext_vector_type(2))) float v2f;
typedef __attribute__((ext_vector_type(8))) float v8f;

// ---------------------------------------------------------------------------
// Per-node projection via V_WMMA_F32_16X16X4_F32 (exact f32 math).
//   out[node][k*3+h] = sum_d vec[node][d][k] * W[d][h]
// One wave (32 lanes) handles 16 nodes:
//   D[16x16] = A[16x4] x B[4x16] accumulated over 4 chunks of d.
//   A[M=h][K=dq]   = W[d][h]              (rows h>=3 are zero, via select)
//   B[K=dq][N=nd]  = vec[node nd][d][k]   (one accumulator per k=0..2)
// f32 16x4 operand layout: VGPR0 -> K=0 (lanes 0-15), K=2 (lanes 16-31);
//                          VGPR1 -> K=1, K=3.  C/D: VGPR r, lanes 0-15 -> M=r.
// A-tile is built branch-free (clamped address + select) so EXEC stays all-1s
// through the entire WMMA body; all W loads issue as a single clause up front.
// ---------------------------------------------------------------------------
__global__ __launch_bounds__(32)
void proj_wmma_kernel(const float* __restrict__ vec,   // [n,16,3]
                      const float* __restrict__ W,     // [16,3]
                      float* __restrict__ out,         // [n,9] (k*3+h)
                      int n) {
  const int lane = threadIdx.x;      // 0..31
  const int col  = lane & 15;        // node column / M row
  const int hi   = lane >> 4;        // half-wave select
  const int node = blockIdx.x * 16 + col;
  const int nodec = node < n ? node : (n - 1);
  const float* __restrict__ src = vec + (long)nodec * 48;

  // ---- A tiles for all 4 chunks, built up front, branch-free ----
  const int  wcol  = col < 3 ? col : 2;          // always-valid address
  const bool valid = (col < 3);
  v2f A[4];
#pragma unroll
  for (int c = 0; c < 4; ++c) {
#pragma unroll
    for (int v = 0; v < 2; ++v) {
      const int d = 4 * c + v + 2 * hi;
      const float w = W[d * 3 + wcol];           // unconditional load
      A[c][v] = valid ? w : 0.0f;                // v_cndmask, EXEC untouched
    }
  }

  v8f acc0 = {}, acc1 = {}, acc2 = {};
#pragma unroll
  for (int c = 0; c < 4; ++c) {
    v2f B0, B1, B2;
#pragma unroll
    for (int v = 0; v < 2; ++v) {
      const int d = 4 * c + v + 2 * hi;
      B0[v] = src[d * 3 + 0];
      B1[v] = src[d * 3 + 1];
      B2[v] = src[d * 3 + 2];
    }
    acc0 = __builtin_amdgcn_wmma_f32_16x16x4_f32(false, A[c], false, B0, (short)0, acc0, false, false);
    acc1 = __builtin_amdgcn_wmma_f32_16x16x4_f32(false, A[c], false, B1, (short)0, acc1, false, false);
    acc2 = __builtin_amdgcn_wmma_f32_16x16x4_f32(false, A[c], false, B2, (short)0, acc2, false, false);
  }
  if (lane < 16 && node < n) {
    float* __restrict__ o = out + (long)node * 9;
#pragma unroll
    for (int h = 0; h < 3; ++h) {
      o[0 * 3 + h] = acc0[h];   // k = 0
      o[1 * 3 + h] = acc1[h];   // k = 1
      o[2 * 3 + h] = acc2[h];   // k = 2
    }
  }
}

// --------------------- order-preserving float<->uint keys -------------------
__device__ __forceinline__ unsigned f2key(float x) {
  unsigned u = __float_as_uint(x);
  return (u & 0x80000000u) ? ~u : (u | 0x80000000u);
}
__device__ __forceinline__ float key2f(unsigned u) {
  return (u & 0x80000000u) ? __uint_as_float(u ^ 0x80000000u)
                           : __uint_as_float(~u);
}

__global__ void init_seg(unsigned* __restrict__ segmax,
                         float* __restrict__ segsum, int n) {
  int i = blockIdx.x * blockDim.x + threadIdx.x;
  if (i < n) { segmax[i] = 0u; segsum[i] = 0.0f; }
}

// ------------------- pass A: raw logit per edge + segment max ---------------
__global__ void edge_raw_kernel(const float* __restrict__ FV,   // [nf,9]
                                const float* __restrict__ TV,   // [nt,9]
                                const float* __restrict__ FF,   // [nf,3,3]
                                const float* __restrict__ TF,   // [nt,3,3]
                                const float* __restrict__ FP,   // [nf,3]
                                const float* __restrict__ TP,   // [nt,3]
                                const float* __restrict__ Wa,   // [24]
                                const int* __restrict__ idx,    // [2,E]
                                int E,
                                float* __restrict__ rawbuf,
                                unsigned* __restrict__ segmax) {
  int e = blockIdx.x * blockDim.x + threadIdx.x;
  if (e >= E) return;
  const int f = idx[e];
  const int t = idx[E + e];

  float fv9[9], tv9[9], ffm[9], tfm[9];
#pragma unroll
  for (int i = 0; i < 9; ++i) {
    fv9[i] = FV[(long)f * 9 + i];
    tv9[i] = TV[(long)t * 9 + i];
    ffm[i] = FF[(long)f * 9 + i];
    tfm[i] = TF[(long)t * 9 + i];
  }
  float pd[3];
#pragma unroll
  for (int k = 0; k < 3; ++k)
    pd[k] = TP[(long)t * 3 + k] - FP[(long)f * 3 + k];

  float v[24];
  // fs rows h=0..2: sum_k FV[k*3+h] * from_frame[k][j]
#pragma unroll
  for (int h = 0; h < 3; ++h)
#pragma unroll
    for (int j = 0; j < 3; ++j)
      v[h * 3 + j] = fv9[h] * ffm[j] + fv9[3 + h] * ffm[3 + j] + fv9[6 + h] * ffm[6 + j];
  // fs row h=3: pos_diff . from_frame[:,j]
#pragma unroll
  for (int j = 0; j < 3; ++j)
    v[9 + j] = pd[0] * ffm[j] + pd[1] * ffm[3 + j] + pd[2] * ffm[6 + j];
  // ts rows h=0..2
#pragma unroll
  for (int h = 0; h < 3; ++h)
#pragma unroll
    for (int j = 0; j < 3; ++j)
      v[12 + h * 3 + j] = tv9[h] * tfm[j] + tv9[3 + h] * tfm[3 + j] + tv9[6 + h] * tfm[6 + j];
  // ts row h=3: (-pos_diff) . to_frame[:,j]
#pragma unroll
  for (int j = 0; j < 3; ++j)
    v[21 + j] = -(pd[0] * tfm[j] + pd[1] * tfm[3 + j] + pd[2] * tfm[6 + j]);

  float r = 0.0f;
#pragma unroll
  for (int i = 0; i < 24; ++i) {
    const float x = v[i];
    const float s = x * (1.0f / (1.0f + __expf(-x)));   // SiLU
    r = fmaf(s, Wa[i], r);
  }
  rawbuf[e] = r;
  atomicMax(&segmax[f], f2key(r));
}

// ------------------- pass B: exp(raw - max) + segment sum -------------------
__global__ void edge_exp_kernel(const int* __restrict__ idx, int E,
                                float* __restrict__ rawbuf,
                                const unsigned* __restrict__ segmax,
                                float* __restrict__ segsum) {
  int e = blockIdx.x * blockDim.x + threadIdx.x;
  if (e >= E) return;
  const int f = idx[e];
  const float ex = __expf(rawbuf[e] - key2f(segmax[f]));
  rawbuf[e] = ex;                     // reuse buffer in place
  atomicAdd(&segsum[f], ex);
}

// ------------------- pass C: normalize --------------------------------------
__global__ void edge_norm_kernel(const int* __restrict__ idx, int E,
                                 const float* __restrict__ rawbuf,
                                 const float* __restrict__ segsum,
                                 float* __restrict__ out) {
  int e = blockIdx.x * blockDim.x + threadIdx.x;
  if (e >= E) return;
  out[e] = rawbuf[e] / segsum[idx[e]];
}

extern "C" void kernel_launch(void* const* d_in, const int* in_sizes, int n_in,
                              void* d_out, int out_size, void* d_ws, size_t ws_size,
                              hipStream_t stream) {
  (void)n_in; (void)out_size; (void)ws_size;
  const float* from_vec   = (const float*)d_in[0];
  const float* to_vec     = (const float*)d_in[1];
  const float* from_frame = (const float*)d_in[2];
  const float* to_frame   = (const float*)d_in[3];
  const float* from_pos   = (const float*)d_in[4];
  const float* to_pos     = (const float*)d_in[5];
  const float* W_from     = (const float*)d_in[6];
  const float* W_to       = (const float*)d_in[7];
  const float* W_attn     = (const float*)d_in[8];
  const int*   inc        = (const int*)d_in[9];

  const int n_from = in_sizes[0] / 48;   // [N_FROM,16,3]
  const int n_to   = in_sizes[1] / 48;   // [N_TO,16,3]
  const int E      = in_sizes[9] / 2;    // [2,E]

  float*    FV     = (float*)d_ws;                       // n_from*9
  float*    TV     = FV + (size_t)n_from * 9;            // n_to*9
  float*    rawbuf = TV + (size_t)n_to * 9;              // E
  unsigned* segmax = (unsigned*)(rawbuf + (size_t)E);    // n_from
  float*    segsum = (float*)(segmax + (size_t)n_from);  // n_from

  proj_wmma_kernel<<<(n_from + 15) / 16, 32, 0, stream>>>(from_vec, W_from, FV, n_from);
  proj_wmma_kernel<<<(n_to + 15) / 16, 32, 0, stream>>>(to_vec, W_to, TV, n_to);
  init_seg<<<(n_from + 255) / 256, 256, 0, stream>>>(segmax, segsum, n_from);

  const int TB = 256;
  edge_raw_kernel<<<(E + TB - 1) / TB, TB, 0, stream>>>(
      FV, TV, from_frame, to_frame, from_pos, to_pos, W_attn, inc, E,
      rawbuf, segmax);
  edge_exp_kernel<<<(E + TB - 1) / TB, TB, 0, stream>>>(inc, E, rawbuf, segmax, segsum);
  edge_norm_kernel<<<(E + TB - 1) / TB, TB, 0, stream>>>(inc, E, rawbuf, segsum, (float*)d_out);
}